// InterpolatorMask_3564822856114
// MI455X (gfx1250) — compile-verified
//
#include <hip/hip_runtime.h>
#include <hip/hip_bf16.h>
#include <math.h>

// CDNA5 (gfx1250) wave32 WMMA: A is 16x4 f32 (2 VGPRs/lane), B is 4x16 f32
// (2 VGPRs/lane), C/D is 16x16 f32 (8 VGPRs/lane).
typedef __attribute__((ext_vector_type(2))) float v2f;
typedef __attribute__((ext_vector_type(8))) float v8f;

// Computes out = valid ? sum_{t=0}^{63} mask[t] * y[(ind+t) % n] : 0
// via a single V_WMMA_F32_16X16X4_F32:
//   A[i][k] = mask[4i+k], B[k][j] = y[(ind + 4j + k) % n]
//   => trace(A*B) = the 64-term windowed dot (exact fp32; terms t>=2 are *0).
__global__ __launch_bounds__(32)
void InterpolatorMask_wmma_kernel(const float* __restrict__ xp,
                                  const float* __restrict__ xOrig,
                                  const float* __restrict__ yOrig,
                                  const float* __restrict__ mask,
                                  float* __restrict__ out,
                                  int n) {
    const int lane = (int)(threadIdx.x & 31u);

    const float x    = xp[0];
    const float x0   = xOrig[0];
    const float dx   = xOrig[1] - x0;
    const float xMax = xOrig[n - 1];

    // ind = floor((x - x0)/dx), then wrap like jnp.roll
    int ind  = (int)floorf((x - x0) / dx);
    int indw = ind % n;
    if (indw < 0) indw += n;

    // Per-lane element indices for BOTH A and B (layouts are symmetric):
    //   lane L holds matrix elements t0 and t0+1, t0 = 4*(L%16) + 2*(L/16)
    const int m  = lane & 15;
    const int h  = lane >> 4;
    const int t0 = 4 * m + 2 * h;

    int iy0 = indw + t0;
    if (iy0 >= n) iy0 -= n;
    int iy1 = indw + t0 + 1;
    if (iy1 >= n) iy1 -= n;

    v2f a, b;
    a.x = mask[t0];
    a.y = mask[t0 + 1];
    b.x = yOrig[iy0];
    b.y = yOrig[iy1];

    v8f c = {};
    // 8 args: (neg_a, A, neg_b, B, c_mod, C, reuse_a, reuse_b)
    c = __builtin_amdgcn_wmma_f32_16x16x4_f32(
            /*neg_a=*/false, a, /*neg_b=*/false, b,
            /*c_mod=*/(short)0, c, /*reuse_a=*/false, /*reuse_b=*/false);

    // Diagonal extraction (D layout: vgpr v, lanes 0-15 -> M=v, N=lane;
    // lanes 16-31 -> M=v+8, N=lane-16):
    //   C[v][v]     lives in lane v,     vgpr v   (v = 0..7)
    //   C[v+8][v+8] lives in lane v+24,  vgpr v   (v = 0..7)
    float r = 0.0f;
#pragma unroll
    for (int v = 0; v < 8; ++v) {
        float cv = c[v];
        r += ((lane == v) || (lane == v + 24)) ? cv : 0.0f;
    }

    // wave32 butterfly reduction -> trace in every lane
#pragma unroll
    for (int off = 16; off > 0; off >>= 1)
        r += __shfl_xor(r, off, 32);

    const bool valid = (x >= x0) && (x < xMax);
    if (lane == 0)
        out[0] = valid ? r : 0.0f;
}

extern "C" void kernel_launch(void* const* d_in, const int* in_sizes, int n_in,
                              void* d_out, int out_size, void* d_ws, size_t ws_size,
                              hipStream_t stream) {
    // setup_inputs order: x (scalar), xOrig, yOrig, mask  (all float32)
    const float* x     = (const float*)d_in[0];
    const float* xOrig = (const float*)d_in[1];
    const float* yOrig = (const float*)d_in[2];
    const float* mask  = (const float*)d_in[3];
    float*       out   = (float*)d_out;
    const int    n     = in_sizes[1];

    // Single wave: the whole windowed dot is one WMMA.
    InterpolatorMask_wmma_kernel<<<1, 32, 0, stream>>>(x, xOrig, yOrig, mask, out, n);
}